// Linear_Attention_26895085208066
// MI455X (gfx1250) — compile-verified
//
#include <hip/hip_runtime.h>
#include <hip/hip_bf16.h>

typedef __attribute__((ext_vector_type(2))) float v2f;
typedef __attribute__((ext_vector_type(8))) float v8f;

__device__ __forceinline__ v8f wmma4(v2f a, v2f b, v8f c) {
    // D = A(16x4) * B(4x16) + C(16x16), fp32
    return __builtin_amdgcn_wmma_f32_16x16x4_f32(
        /*neg_a=*/false, a, /*neg_b=*/false, b,
        /*c_mod=*/(short)0, c, /*reuse_a=*/false, /*reuse_b=*/false);
}

// ---------------------------------------------------------------------------
// NT GEMM: C[m,n] = sum_k A[m,k] * W[n,k]   (A: MxK row-major, W: NxK row-major)
// 128x128 block tile, 8 waves of 32x64, K staged 32 at a time in double-
// buffered LDS with register prefetch (one barrier per K-tile).
// act: 0 = none, 1 = elu(x)+1. split: 0 -> flat MxN (+bias), 1 -> (B,H,T,d).
// Fixed shapes: M % 128 == 0, N % 128 == 0, K % 32 == 0, T=2048, H=16, d=64.
// ---------------------------------------------------------------------------
#define G_TM 128
#define G_TN 128
#define G_TK 32
#define G_PAD 36   // 36*4B stride: 16B-aligned float4 stores, conflict-free reads

__global__ __launch_bounds__(256) void gemm_nt_kernel(
    const float* __restrict__ A, const float* __restrict__ W,
    const float* __restrict__ bias, float* __restrict__ C,
    int M, int N, int K, int act, int split)
{
    __shared__ float As[2][G_TM * G_PAD];
    __shared__ float Bs[2][G_TN * G_PAD];

    const int tid   = threadIdx.x;
    const int lane  = tid & 31;
    const int lrow  = lane & 15;          // fragment row/col within 16
    const int khalf = (lane >> 4) * 2;    // K sub-offset per ISA 32-bit A/B layout
    const int rhalf = (lane >> 4) * 8;    // C-row sub-offset per ISA C layout
    const int wave  = tid >> 5;

    const int m0 = blockIdx.x * G_TM;
    const int n0 = blockIdx.y * G_TN;
    const int wm = (wave & 3) * 32;       // wave's 32-row strip inside tile
    const int wn = (wave >> 2) * 64;      // wave's 64-col strip inside tile

    // cooperative-load geometry: 128x32 floats = 1024 float4 / 256 threads = 4 each
    const int ldRow = tid >> 3;           // 0..31 (+32*i)
    const int ldCol = (tid & 7) * 4;      // 0..28

    v8f cacc[2][4] = {};
    float4 pa[4], pb[4];

    // ---- prefetch helpers ----
    auto loadTile = [&](int kt) {
#pragma unroll
        for (int i = 0; i < 4; ++i)
            pa[i] = *(const float4*)&A[(size_t)(m0 + ldRow + 32 * i) * K + kt + ldCol];
#pragma unroll
        for (int i = 0; i < 4; ++i)
            pb[i] = *(const float4*)&W[(size_t)(n0 + ldRow + 32 * i) * K + kt + ldCol];
    };
    auto storeTile = [&](int p) {
#pragma unroll
        for (int i = 0; i < 4; ++i)
            *(float4*)&As[p][(ldRow + 32 * i) * G_PAD + ldCol] = pa[i];
#pragma unroll
        for (int i = 0; i < 4; ++i)
            *(float4*)&Bs[p][(ldRow + 32 * i) * G_PAD + ldCol] = pb[i];
    };

    loadTile(0);
    storeTile(0);
    __syncthreads();

    int p = 0;
    for (int kt = 0; kt < K; kt += G_TK) {
        const bool hasNext = (kt + G_TK) < K;
        if (hasNext) loadTile(kt + G_TK);   // global -> regs, overlapped with MACs

#pragma unroll
        for (int kk = 0; kk < G_TK; kk += 4) {
            v2f af[2], bf[4];
            af[0] = *(const v2f*)&As[p][(wm +      lrow) * G_PAD + kk + khalf];
            af[1] = *(const v2f*)&As[p][(wm + 16 + lrow) * G_PAD + kk + khalf];
#pragma unroll
            for (int fn = 0; fn < 4; ++fn)
                bf[fn] = *(const v2f*)&Bs[p][(wn + fn * 16 + lrow) * G_PAD + kk + khalf];
#pragma unroll
            for (int fm = 0; fm < 2; ++fm)
#pragma unroll
                for (int fn = 0; fn < 4; ++fn)
                    cacc[fm][fn] = wmma4(af[fm], bf[fn], cacc[fm][fn]);
        }

        if (hasNext) storeTile(p ^ 1);      // regs -> other LDS buffer
        __syncthreads();
        p ^= 1;
    }

    // Epilogue
#pragma unroll
    for (int fm = 0; fm < 2; ++fm)
#pragma unroll
        for (int fn = 0; fn < 4; ++fn)
#pragma unroll
            for (int r = 0; r < 8; ++r) {
                int row = m0 + wm + fm * 16 + r + rhalf;
                int col = n0 + wn + fn * 16 + lrow;
                float v = cacc[fm][fn][r];
                if (act == 1) v = (v > 0.0f) ? (v + 1.0f) : __expf(v);
                if (bias) v += bias[col];
                if (split) {
                    // (B,H,T,d): row = b*2048 + t ; col = h*64 + dd
                    int bb = row >> 11, t = row & 2047;
                    int h  = col >> 6,  dd = col & 63;
                    C[(((size_t)bb * 16 + h) * 2048 + t) * 64 + dd] = v;
                } else {
                    C[(size_t)row * N + col] = v;
                }
            }
}

// ---------------------------------------------------------------------------
// Chunked linear-attention scan. One workgroup (256 threads, 8 waves) per
// (b,h). Carries S (64x64) and z (64) in LDS across the 16 chunks of 128.
// Q,K,V are (B,H,T,d) with phi already applied to Q,K. O is (B,T,E) flat.
// ---------------------------------------------------------------------------
#define C_PAD  68    // q/k/v row stride (floats): 16B-aligned float4 stores
#define A_PAD  130   // masked-A row stride
#define S_PAD  66    // S row stride

__global__ __launch_bounds__(256) void linattn_scan_kernel(
    const float* __restrict__ Q, const float* __restrict__ Kx,
    const float* __restrict__ V, float* __restrict__ O)
{
    __shared__ float qs[128 * C_PAD];
    __shared__ float ks[128 * C_PAD];
    __shared__ float vs[128 * C_PAD];
    __shared__ float Am[128 * A_PAD];
    __shared__ float Sm[64 * S_PAD];
    __shared__ float zv[64];
    __shared__ float den[128];

    const int tid   = threadIdx.x;
    const int lane  = tid & 31;
    const int lrow  = lane & 15;
    const int khalf = (lane >> 4) * 2;
    const int rhalf = (lane >> 4) * 8;
    const int wave  = tid >> 5;

    const int bh = blockIdx.x;            // b*16 + h
    const int b  = bh >> 4, h = bh & 15;
    const size_t base = (size_t)bh * 2048 * 64;

    for (int i = tid; i < 64 * S_PAD; i += 256) Sm[i] = 0.0f;
    if (tid < 64) zv[tid] = 0.0f;
    __syncthreads();

    for (int jc = 0; jc < 16; ++jc) {
        const size_t cbase = base + (size_t)jc * 128 * 64;

        // ---- Stage q,k,v chunk (128x64 each) into LDS ----
#pragma unroll
        for (int i = 0; i < 8; ++i) {
            int idx = tid + i * 256;              // float4 index within 128x16
            int row = idx >> 4, c4 = (idx & 15) * 4;
            *(float4*)&qs[row * C_PAD + c4] = *(const float4*)&Q [cbase + row * 64 + c4];
            *(float4*)&ks[row * C_PAD + c4] = *(const float4*)&Kx[cbase + row * 64 + c4];
            *(float4*)&vs[row * C_PAD + c4] = *(const float4*)&V [cbase + row * 64 + c4];
        }
        __syncthreads();

        // ---- A = q k^T (128x128, K=64), causal-masked, into Am ----
        {
            const int m0 = wave * 16;             // 8 waves x 16 rows
            v8f acc[8] = {};
            for (int kk = 0; kk < 64; kk += 4) {
                v2f a = *(const v2f*)&qs[(m0 + lrow) * C_PAD + kk + khalf];
#pragma unroll
                for (int c = 0; c < 8; ++c) {
                    v2f bf = *(const v2f*)&ks[(c * 16 + lrow) * C_PAD + kk + khalf];
                    acc[c] = wmma4(a, bf, acc[c]);
                }
            }
#pragma unroll
            for (int c = 0; c < 8; ++c)
#pragma unroll
                for (int r = 0; r < 8; ++r) {
                    int row = m0 + r + rhalf;
                    int col = c * 16 + lrow;
                    Am[row * A_PAD + col] = (row >= col) ? acc[c][r] : 0.0f;
                }
        }
        __syncthreads();

        // ---- den[i] = eps + rowsum(Amask) + q[i]·z ----
        if (tid < 128) {
            float s = 1e-5f;
            for (int j = 0; j <= tid; ++j) s += Am[tid * A_PAD + j];
            float qz = 0.0f;
            for (int k = 0; k < 64; ++k) qz += qs[tid * C_PAD + k] * zv[k];
            den[tid] = s + qz;
        }
        __syncthreads();

        // ---- num = Amask @ v (K=128)  +  q @ S (K=64) ; out = num/den ----
        {
            const int m0 = wave * 16;
            v8f acc[4] = {};
            for (int kk = 0; kk < 128; kk += 4) {
                v2f a = *(const v2f*)&Am[(m0 + lrow) * A_PAD + kk + khalf];
#pragma unroll
                for (int c = 0; c < 4; ++c) {
                    v2f bf;
                    bf.x = vs[(kk + khalf)     * C_PAD + c * 16 + lrow];
                    bf.y = vs[(kk + khalf + 1) * C_PAD + c * 16 + lrow];
                    acc[c] = wmma4(a, bf, acc[c]);
                }
            }
            for (int kk = 0; kk < 64; kk += 4) {
                v2f a = *(const v2f*)&qs[(m0 + lrow) * C_PAD + kk + khalf];
#pragma unroll
                for (int c = 0; c < 4; ++c) {
                    v2f bf;
                    bf.x = Sm[(kk + khalf)     * S_PAD + c * 16 + lrow];
                    bf.y = Sm[(kk + khalf + 1) * S_PAD + c * 16 + lrow];
                    acc[c] = wmma4(a, bf, acc[c]);
                }
            }
#pragma unroll
            for (int c = 0; c < 4; ++c)
#pragma unroll
                for (int r = 0; r < 8; ++r) {
                    int row = m0 + r + rhalf;
                    int col = c * 16 + lrow;
                    float o = acc[c][r] / den[row];
                    O[((size_t)(b * 2048 + jc * 128 + row)) * 1024 + h * 64 + col] = o;
                }
        }
        __syncthreads();   // all reads of Sm complete before update

        // ---- S += k^T v (64x64, K=128): wave -> tile row (wave&3), col pair ----
        {
            const int tr  = (wave & 3) * 16;
            const int tc0 = (wave >> 2) * 32;
            v8f acc[2];
#pragma unroll
            for (int c = 0; c < 2; ++c)
#pragma unroll
                for (int r = 0; r < 8; ++r)
                    acc[c][r] = Sm[(tr + r + rhalf) * S_PAD + tc0 + c * 16 + lrow];
            for (int kk = 0; kk < 128; kk += 4) {
                v2f a;
                a.x = ks[(kk + khalf)     * C_PAD + tr + lrow];
                a.y = ks[(kk + khalf + 1) * C_PAD + tr + lrow];
#pragma unroll
                for (int c = 0; c < 2; ++c) {
                    v2f bf;
                    bf.x = vs[(kk + khalf)     * C_PAD + tc0 + c * 16 + lrow];
                    bf.y = vs[(kk + khalf + 1) * C_PAD + tc0 + c * 16 + lrow];
                    acc[c] = wmma4(a, bf, acc[c]);
                }
            }
#pragma unroll
            for (int c = 0; c < 2; ++c)
#pragma unroll
                for (int r = 0; r < 8; ++r)
                    Sm[(tr + r + rhalf) * S_PAD + tc0 + c * 16 + lrow] = acc[c][r];
        }
        __syncthreads();

        // ---- z += colsum(k) ----
        if (tid < 64) {
            float s = zv[tid];
            for (int i = 0; i < 128; ++i) s += ks[i * C_PAD + tid];
            zv[tid] = s;
        }
        __syncthreads();
    }
}

extern "C" void kernel_launch(void* const* d_in, const int* in_sizes, int n_in,
                              void* d_out, int out_size, void* d_ws, size_t ws_size,
                              hipStream_t stream) {
    const float* x  = (const float*)d_in[0];
    const float* Wq = (const float*)d_in[1];
    const float* Wk = (const float*)d_in[2];
    const float* Wv = (const float*)d_in[3];
    const float* Wo = (const float*)d_in[4];
    const float* bo = (const float*)d_in[5];

    const int T = 2048, E = 1024;
    const int B = in_sizes[0] / (T * E);   // == 2
    const int M = B * T;                   // 4096
    const size_t plane = (size_t)M * E;

    float* phiQ = (float*)d_ws;            // (B,H,T,d)
    float* phiK = phiQ + plane;            // (B,H,T,d)
    float* Vh   = phiK + plane;            // (B,H,T,d)
    float* attn = Vh   + plane;            // (B,T,E) flat

    dim3 blk(256);
    dim3 gg(M / G_TM, E / G_TN);

    // Projections (phi = elu+1 fused for Q,K), head-split layout
    gemm_nt_kernel<<<gg, blk, 0, stream>>>(x, Wq, nullptr, phiQ, M, E, E, /*act=*/1, /*split=*/1);
    gemm_nt_kernel<<<gg, blk, 0, stream>>>(x, Wk, nullptr, phiK, M, E, E, /*act=*/1, /*split=*/1);
    gemm_nt_kernel<<<gg, blk, 0, stream>>>(x, Wv, nullptr, Vh,   M, E, E, /*act=*/0, /*split=*/1);

    // Chunked recurrent scan, one WG per (b,h)
    linattn_scan_kernel<<<dim3(B * 16), blk, 0, stream>>>(phiQ, phiK, Vh, attn);

    // Output projection + bias
    gemm_nt_kernel<<<gg, blk, 0, stream>>>(attn, Wo, bo, (float*)d_out, M, E, E, /*act=*/0, /*split=*/0);
}